// MassEmbeddingLayer_7215545057629
// MI455X (gfx1250) — compile-verified
//
#include <hip/hip_runtime.h>

typedef __attribute__((ext_vector_type(2))) float v2f;
typedef __attribute__((ext_vector_type(8))) float v8f;

#define NROWS 8192
#define DIM   64
#define NTILE (NROWS / 16)   // 512 column tiles

// ---------------------------------------------------------------------------
// Kernel 1: emb = x*w + b  (IN_DIM==1), write emb to out, xn (row-normalized)
// to workspace, zero the aux-loss accumulator.
// ---------------------------------------------------------------------------
__global__ __launch_bounds__(256) void emb_xn_kernel(
    const float* __restrict__ x, const float* __restrict__ w,
    const float* __restrict__ b, float* __restrict__ emb,
    float* __restrict__ xn, float* __restrict__ aux) {
  const int i = blockIdx.x * blockDim.x + threadIdx.x;
  if (i == 0) aux[0] = 0.0f;  // accumulator for kernel 2 (stream-ordered)
  if (i >= NROWS) return;

  const float xi = x[i];
  float e[DIM];
  float ss = 0.0f;
#pragma unroll
  for (int j = 0; j < DIM; ++j) {
    const float v = fmaf(xi, w[j], b[j]);
    e[j] = v;
    ss = fmaf(v, v, ss);
  }
  const float invn = 1.0f / sqrtf(fmaxf(ss, 1e-12f));
#pragma unroll
  for (int j = 0; j < DIM; ++j) {
    emb[(size_t)i * DIM + j] = e[j];
    xn[(size_t)i * DIM + j] = e[j] * invn;
  }
}

// ---------------------------------------------------------------------------
// Kernel 2: per-row min_j |xn[i]·xn[j]| via V_WMMA_F32_16X16X4_F32,
// then aux += -(1 - min)/N  (== ALPHA * (-mean(max_j(1-|cos|)))).
// One wave32 per block; each wave owns two 16-row M tiles (32 rows).
// ---------------------------------------------------------------------------
__global__ __launch_bounds__(32) void cos_min_kernel(
    const float* __restrict__ xn, float* __restrict__ aux) {
  const int lane = threadIdx.x;   // 0..31 (wave32)
  const int lo = lane & 15;
  const int hi = lane >> 4;       // 0 or 1
  const int m0 = blockIdx.x * 32; // rows [m0, m0+32)

  // Preload A fragments for both M tiles, all 16 K-steps (K = 4 per WMMA).
  // f32 16x4 A layout: lanes 0-15 hold K={0,1}, lanes 16-31 hold K={2,3}.
  const float* rowA0 = xn + (size_t)(m0 + lo) * DIM + 2 * hi;
  const float* rowA1 = rowA0 + (size_t)16 * DIM;
  v2f a0[16], a1[16];
#pragma unroll
  for (int k = 0; k < 16; ++k) {
    a0[k] = *(const v2f*)(rowA0 + 4 * k);
    a1[k] = *(const v2f*)(rowA1 + 4 * k);
  }

  v8f mn0, mn1;
#pragma unroll
  for (int r = 0; r < 8; ++r) { mn0[r] = 3.0e38f; mn1[r] = 3.0e38f; }

  for (int jt = 0; jt < NTILE; ++jt) {
    // B[k][n] = xn[jt*16 + n][k] -> identical per-lane addressing as A.
    const float* rowB = xn + (size_t)(jt * 16 + lo) * DIM + 2 * hi;
    if (jt + 1 < NTILE)
      __builtin_prefetch(rowB + (size_t)16 * DIM, 0, 1);  // global_prefetch_b8

    v2f bf[16];
#pragma unroll
    for (int k = 0; k < 16; ++k) bf[k] = *(const v2f*)(rowB + 4 * k);

    v8f c0 = {}, c1 = {};
#pragma unroll
    for (int k = 0; k < 16; ++k) {
      c0 = __builtin_amdgcn_wmma_f32_16x16x4_f32(false, a0[k], false, bf[k],
                                                 (short)0, c0, false, false);
      c1 = __builtin_amdgcn_wmma_f32_16x16x4_f32(false, a1[k], false, bf[k],
                                                 (short)0, c1, false, false);
    }
#pragma unroll
    for (int r = 0; r < 8; ++r) {
      mn0[r] = fminf(mn0[r], fabsf(c0[r]));
      mn1[r] = fminf(mn1[r], fabsf(c1[r]));
    }
  }

  // C layout: VGPR r holds row (r | r+8) at column class (lane&15).
  // Min-reduce across each 16-lane group, accumulate sum of (1 - rowmin).
  float s = 0.0f;
#pragma unroll
  for (int r = 0; r < 8; ++r) {
    float v0 = mn0[r], v1 = mn1[r];
#pragma unroll
    for (int m = 8; m >= 1; m >>= 1) {
      v0 = fminf(v0, __shfl_xor(v0, m, 32));
      v1 = fminf(v1, __shfl_xor(v1, m, 32));
    }
    s += (1.0f - v0) + (1.0f - v1);
  }
  // lanes 0-15 summed rows {m0..+7, m0+16..+23}; lanes 16-31 the others.
  s += __shfl_xor(s, 16, 32);
  if (lane == 0) atomicAdd(aux, -s * (1.0f / (float)NROWS));
}

// ---------------------------------------------------------------------------
extern "C" void kernel_launch(void* const* d_in, const int* in_sizes, int n_in,
                              void* d_out, int out_size, void* d_ws,
                              size_t ws_size, hipStream_t stream) {
  const float* x = (const float*)d_in[0];  // [8192,1]
  const float* w = (const float*)d_in[1];  // [1,64]
  const float* b = (const float*)d_in[2];  // [64]
  float* out = (float*)d_out;
  float* emb = out;                         // 8192*64 floats
  float* aux = out + (size_t)NROWS * DIM;   // 1 float
  float* xn = (float*)d_ws;                 // 8192*64 floats = 2 MB

  emb_xn_kernel<<<NROWS / 256, 256, 0, stream>>>(x, w, b, emb, xn, aux);
  cos_min_kernel<<<NROWS / 32, 32, 0, stream>>>(xn, aux);
}